// TextCNN2_88897233092837
// MI455X (gfx1250) — compile-verified
//
#include <hip/hip_runtime.h>
#include <hip/hip_bf16.h>

typedef __attribute__((ext_vector_type(2))) float v2f;
typedef __attribute__((ext_vector_type(8))) float v8f;

#define E 300
#define V_TOK 50000
#define S_LEN 512
#define NROW 16   // 14 used weight rows, padded to 16

// ---------------------------------------------------------------------------
// Phase 0: pack the 14 conv weight rows into K-major Wt[k*16 + j], j padded to 16.
// j: 0..3 = w1 rows, 4..6 = w3, 7..9 = w4, 10..11 = w5, 12..13 = w6, 14..15 = 0
// ---------------------------------------------------------------------------
__global__ void pack_weights(const float* __restrict__ w1, const float* __restrict__ w3,
                             const float* __restrict__ w4, const float* __restrict__ w5,
                             const float* __restrict__ w6, float* __restrict__ Wt) {
    int idx = blockIdx.x * blockDim.x + threadIdx.x;
    if (idx >= E * NROW) return;
    int k = idx >> 4;
    int j = idx & 15;
    float v = 0.0f;
    if (j < 4)       v = w1[j * E + k];
    else if (j < 7)  v = w3[(j - 4) * E + k];
    else if (j < 10) v = w4[(j - 7) * E + k];
    else if (j < 12) v = w5[(j - 10) * E + k];
    else if (j < 14) v = w6[(j - 12) * E + k];
    Wt[idx] = v;
}

// ---------------------------------------------------------------------------
// Phase 1: P[v][j] = dot(emb[v], wrow_j) via V_WMMA_F32_16X16X4_F32.
// One wave per 16-row tile of the embedding table. 75 WMMA steps (K=300).
// A 16x4 f32: lane<16 -> K={0,1}, lane>=16 -> K={2,3}; B mirrored with N on lanes.
// ---------------------------------------------------------------------------
__global__ void wmma_proj(const float* __restrict__ emb, const float* __restrict__ Wt,
                          float* __restrict__ P) {
    const int lane = threadIdx.x;             // 0..31, one wave per block
    const int m0   = blockIdx.x * 16;
    const int mr   = lane & 15;
    const int k0   = (lane >> 4) << 1;        // 0 or 2

    const float* __restrict__ arow = emb + (size_t)(m0 + mr) * E + k0;
    v8f c = {0.f, 0.f, 0.f, 0.f, 0.f, 0.f, 0.f, 0.f};

    for (int kc = 0; kc < E; kc += 4) {
        v2f a, b;
        a.x = arow[kc];
        a.y = arow[kc + 1];
        b.x = Wt[(kc + k0) * NROW + mr];
        b.y = Wt[(kc + k0 + 1) * NROW + mr];
        // (neg_a, A, neg_b, B, c_mod, C, reuse_a, reuse_b)
        c = __builtin_amdgcn_wmma_f32_16x16x4_f32(false, a, false, b, (short)0, c,
                                                  false, false);
    }

    const int mhi = (lane >> 4) << 3;         // 0 or 8
#pragma unroll
    for (int i = 0; i < 8; ++i) {
        P[(size_t)(m0 + i + mhi) * NROW + mr] = c[i];
    }
}

// ---------------------------------------------------------------------------
// Phase 2: one block per batch. Gather P rows for the 512 tokens into LDS,
// sliding-window sums per conv, block-max, tanh/bias/fc/sigmoid at the end.
// ---------------------------------------------------------------------------
#define SP_PITCH 17   // pad to avoid LDS bank conflicts

__global__ void __launch_bounds__(256) textcnn_reduce(
        const int* __restrict__ x, const float* __restrict__ P,
        const float* __restrict__ b1, const float* __restrict__ b3,
        const float* __restrict__ b4, const float* __restrict__ b5,
        const float* __restrict__ b6, const float* __restrict__ fc_w,
        const float* __restrict__ fc_b, float* __restrict__ out) {
    __shared__ float sp[S_LEN * SP_PITCH];
    __shared__ float red[5][8];

    const int tid = threadIdx.x;
    const int bb  = blockIdx.x;

    // Stage gathered projections: sp[t][j] = P[x[b,t]][j]
    for (int t = tid; t < S_LEN; t += 256) {
        int token = x[bb * S_LEN + t];
        const float4* pr = (const float4*)(P + (size_t)token * NROW);
        float4 r0 = pr[0], r1 = pr[1], r2 = pr[2], r3 = pr[3];
        float* d = &sp[t * SP_PITCH];
        d[0]  = r0.x; d[1]  = r0.y; d[2]  = r0.z; d[3]  = r0.w;
        d[4]  = r1.x; d[5]  = r1.y; d[6]  = r1.z; d[7]  = r1.w;
        d[8]  = r2.x; d[9]  = r2.y; d[10] = r2.z; d[11] = r2.w;
        d[12] = r3.x; d[13] = r3.y;
    }
    __syncthreads();

    float m1 = -INFINITY, m3 = -INFINITY, m4 = -INFINITY;
    float m5 = -INFINITY, m6 = -INFINITY;

    for (int t = tid; t < S_LEN; t += 256) {
        const float* r = &sp[t * SP_PITCH];
        if (t <= S_LEN - 4) {   // conv1, K=4
            float v = r[0] + r[SP_PITCH + 1] + r[2 * SP_PITCH + 2] + r[3 * SP_PITCH + 3];
            m1 = fmaxf(m1, v);
        }
        if (t <= S_LEN - 3) {   // conv3 / conv4, K=3
            m3 = fmaxf(m3, r[4] + r[SP_PITCH + 5] + r[2 * SP_PITCH + 6]);
            m4 = fmaxf(m4, r[7] + r[SP_PITCH + 8] + r[2 * SP_PITCH + 9]);
        }
        if (t <= S_LEN - 2) {   // conv5 / conv6, K=2
            m5 = fmaxf(m5, r[10] + r[SP_PITCH + 11]);
            m6 = fmaxf(m6, r[12] + r[SP_PITCH + 13]);
        }
    }

    // wave32 max-reduce
#pragma unroll
    for (int off = 16; off >= 1; off >>= 1) {
        m1 = fmaxf(m1, __shfl_xor(m1, off, 32));
        m3 = fmaxf(m3, __shfl_xor(m3, off, 32));
        m4 = fmaxf(m4, __shfl_xor(m4, off, 32));
        m5 = fmaxf(m5, __shfl_xor(m5, off, 32));
        m6 = fmaxf(m6, __shfl_xor(m6, off, 32));
    }
    if ((tid & 31) == 0) {
        int w = tid >> 5;
        red[0][w] = m1; red[1][w] = m3; red[2][w] = m4; red[3][w] = m5; red[4][w] = m6;
    }
    __syncthreads();

    if (tid == 0) {
        float M1 = red[0][0], M3 = red[1][0], M4 = red[2][0], M5 = red[3][0], M6 = red[4][0];
#pragma unroll
        for (int w = 1; w < 8; ++w) {
            M1 = fmaxf(M1, red[0][w]); M3 = fmaxf(M3, red[1][w]);
            M4 = fmaxf(M4, red[2][w]); M5 = fmaxf(M5, red[3][w]);
            M6 = fmaxf(M6, red[4][w]);
        }
        // tanh is monotonic: max(tanh(y+b)) == tanh(max(y)+b)
        float o1 = tanhf(M1 + b1[0]);
        float o3 = tanhf(M3 + b3[0]);
        float o4 = tanhf(M4 + b4[0]);
        float o5 = tanhf(M5 + b5[0]);
        float o6 = tanhf(M6 + b6[0]);
        // feature order: [o1, o2(=o1 due to reference bug), o3, o4, o5, o6]
        float logit = fc_b[0] + fc_w[0] * o1 + fc_w[1] * o1 + fc_w[2] * o3 +
                      fc_w[3] * o4 + fc_w[4] * o5 + fc_w[5] * o6;
        out[bb] = 1.0f / (1.0f + expf(-logit));
    }
}

// ---------------------------------------------------------------------------
extern "C" void kernel_launch(void* const* d_in, const int* in_sizes, int n_in,
                              void* d_out, int out_size, void* d_ws, size_t ws_size,
                              hipStream_t stream) {
    const int*   x    = (const int*)d_in[0];
    const float* emb  = (const float*)d_in[1];
    const float* w1   = (const float*)d_in[2];
    const float* b1   = (const float*)d_in[3];
    // d_in[4], d_in[5] = w2, b2 : dead in the reference (conv1 used twice)
    const float* w3   = (const float*)d_in[6];
    const float* b3   = (const float*)d_in[7];
    const float* w4   = (const float*)d_in[8];
    const float* b4   = (const float*)d_in[9];
    const float* w5   = (const float*)d_in[10];
    const float* b5   = (const float*)d_in[11];
    const float* w6   = (const float*)d_in[12];
    const float* b6   = (const float*)d_in[13];
    const float* fc_w = (const float*)d_in[14];
    const float* fc_b = (const float*)d_in[15];
    float* out = (float*)d_out;

    float* Wt = (float*)d_ws;              // 300*16 floats = 19.2 KB
    float* P  = Wt + E * NROW;             // 50000*16 floats = 3.2 MB

    pack_weights<<<dim3((E * NROW + 255) / 256), dim3(256), 0, stream>>>(
        w1, w3, w4, w5, w6, Wt);

    wmma_proj<<<dim3(V_TOK / 16), dim3(32), 0, stream>>>(emb, Wt, P);

    textcnn_reduce<<<dim3(256), dim3(256), 0, stream>>>(
        x, P, b1, b3, b4, b5, b6, fc_w, fc_b, out);
}